// GCN3Regressor_35021163332023
// MI455X (gfx1250) — compile-verified
//
#include <hip/hip_runtime.h>
#include <hip/hip_bf16.h>

// ---------------------------------------------------------------------------
// GCN 3-layer regressor for MI455X (gfx1250, wave32).
// GEMMs use V_WMMA_F32_16X16X4_F32 (fp32 WMMA keeps reference precision; the
// big GEMM is HBM-traffic-bound reading x once, not matrix-rate-bound).
// Edge aggregation uses global_atomic_add_f32 into L2-resident buffers.
// ---------------------------------------------------------------------------

typedef float v2f __attribute__((ext_vector_type(2)));
typedef float v8f __attribute__((ext_vector_type(8)));

#define IN_C 512
#define H1C 64
#define H2C 32

// ---- gcn_norm ------------------------------------------------------------

__global__ void k_init_deg(float* __restrict__ deg, int Nn) {
  int i = blockIdx.x * blockDim.x + threadIdx.x;
  if (i < Nn) deg[i] = 1.0f;  // self-loop weight contributes 1 to deg[col=i]
}

__global__ void k_deg_accum(const int* __restrict__ ei, const float* __restrict__ ew,
                            float* __restrict__ deg, int E) {
  int e = blockIdx.x * blockDim.x + threadIdx.x;
  if (e >= E) return;
  int col = ei[E + e];
  atomicAdd(&deg[col], ew[e]);
}

__global__ void k_dinv(float* __restrict__ dinv, int Nn) {
  int i = blockIdx.x * blockDim.x + threadIdx.x;
  if (i >= Nn) return;
  float d = dinv[i];
  dinv[i] = (d > 0.0f) ? rsqrtf(fmaxf(d, 1e-12f)) : 0.0f;
}

__global__ void k_wnorm(const int* __restrict__ ei, const float* __restrict__ ew,
                        const float* __restrict__ dinv, float* __restrict__ wnorm, int E) {
  int e = blockIdx.x * blockDim.x + threadIdx.x;
  if (e >= E) return;
  int row = ei[e];
  int col = ei[E + e];
  wnorm[e] = dinv[row] * ew[e] * dinv[col];
}

// ---- WMMA GEMM: C[N x NOUT] = (relu?)A[N x K] @ B[K x NOUT] ---------------
// One wave per 16x16 C tile; NT waves cover NOUT = NT*16 columns.
// f32 16x16x4 WMMA: A lane l holds A[l%16][2*(l/16)+{0,1}],
//                   B lane l holds B[2*(l/16)+{0,1}][l%16],
//                   C vgpr v  holds C[v+8*(l/16)][l%16].

template <int K, int NT, bool RELU>
__global__ __launch_bounds__(NT * 32) void k_gemm_wmma_f32(
    const float* __restrict__ A, const float* __restrict__ B, float* __restrict__ C) {
  constexpr int NOUT = NT * 16;
  const int wave = threadIdx.x >> 5;
  const int lane = threadIdx.x & 31;
  const int m = lane & 15;
  const int half = lane >> 4;
  const int n0 = wave * 16;
  const long long rowBase = (long long)blockIdx.x * 16;

  const float* arow = A + (rowBase + m) * K + half * 2;  // 8B aligned (K even)

  v8f acc = {};
#pragma unroll 4
  for (int k = 0; k < K; k += 4) {
    float2 av = *reinterpret_cast<const float2*>(arow + k);
    v2f a;
    a.x = RELU ? fmaxf(av.x, 0.0f) : av.x;
    a.y = RELU ? fmaxf(av.y, 0.0f) : av.y;
    const int kr = k + half * 2;
    v2f b;
    b.x = B[kr * NOUT + n0 + m];
    b.y = B[(kr + 1) * NOUT + n0 + m];
    acc = __builtin_amdgcn_wmma_f32_16x16x4_f32(false, a, false, b, (short)0, acc,
                                                false, false);
  }

#pragma unroll
  for (int v = 0; v < 8; ++v) {
    long long r = rowBase + v + 8 * half;
    C[r * NOUT + n0 + m] = acc[v];
  }
}

// ---- aggregation ---------------------------------------------------------
// agg[i][f] = bias[f] + dinv[i]^2 * h[i][f]   (self-loop term + bias)
template <int F>
__global__ void k_init_agg(float* __restrict__ agg, const float* __restrict__ h,
                           const float* __restrict__ dinv, const float* __restrict__ bias,
                           int Nn) {
  long long t = (long long)blockIdx.x * blockDim.x + threadIdx.x;
  if (t >= (long long)Nn * F) return;
  int node = (int)(t / F);
  int f = (int)(t % F);
  float di = dinv[node];
  agg[t] = bias[f] + di * di * h[t];
}

// agg[col] += wnorm[e] * h[row], CH float4-chunks per edge (F = CH*4)
template <int CH>
__global__ __launch_bounds__(256) void k_edge_agg(const int* __restrict__ ei,
                                                  const float* __restrict__ wnorm,
                                                  const float* __restrict__ h,
                                                  float* __restrict__ agg, int E) {
  long long t = (long long)blockIdx.x * blockDim.x + threadIdx.x;
  if (t >= (long long)E * CH) return;
  int e = (int)(t / CH);
  int c = (int)(t % CH);
  int row = ei[e];
  int col = ei[E + e];
  float w = wnorm[e];
  float4 v = reinterpret_cast<const float4*>(h + (size_t)row * (CH * 4))[c];
  float* dst = agg + (size_t)col * (CH * 4) + c * 4;
  atomicAdd(dst + 0, w * v.x);
  atomicAdd(dst + 1, w * v.y);
  atomicAdd(dst + 2, w * v.z);
  atomicAdd(dst + 3, w * v.w);
}

// ---- layer 3 (out dim 1): g[i] = sum_f relu(a2[i][f]) * W3[f] -------------
__global__ void k_gemm3_relu(const float* __restrict__ a2, const float* __restrict__ W3,
                             float* __restrict__ g, int Nn) {
  int i = blockIdx.x * blockDim.x + threadIdx.x;
  if (i >= Nn) return;
  const float* r = a2 + (size_t)i * H2C;
  float acc = 0.0f;
#pragma unroll
  for (int f = 0; f < H2C; ++f) acc += fmaxf(r[f], 0.0f) * W3[f];
  g[i] = acc;
}

__global__ void k_init_out(float* __restrict__ out, const float* __restrict__ g,
                           const float* __restrict__ dinv, const float* __restrict__ b3,
                           int Nn) {
  int i = blockIdx.x * blockDim.x + threadIdx.x;
  if (i >= Nn) return;
  float di = dinv[i];
  out[i] = b3[0] + di * di * g[i];
}

__global__ void k_edge_agg1(const int* __restrict__ ei, const float* __restrict__ wnorm,
                            const float* __restrict__ g, float* __restrict__ out, int E) {
  int e = blockIdx.x * blockDim.x + threadIdx.x;
  if (e >= E) return;
  int row = ei[e];
  int col = ei[E + e];
  atomicAdd(&out[col], wnorm[e] * g[row]);
}

// ---------------------------------------------------------------------------

static inline unsigned int blocks_for(long long total, int bs) {
  return (unsigned int)((total + bs - 1) / bs);
}

extern "C" void kernel_launch(void* const* d_in, const int* in_sizes, int n_in,
                              void* d_out, int out_size, void* d_ws, size_t ws_size,
                              hipStream_t stream) {
  const float* x = (const float*)d_in[0];
  const int* ei = (const int*)d_in[1];  // JAX default x64=off -> int32
  const float* ew = (const float*)d_in[2];
  const float* W1 = (const float*)d_in[3];
  const float* b1 = (const float*)d_in[4];
  const float* W2 = (const float*)d_in[5];
  const float* b2 = (const float*)d_in[6];
  const float* W3 = (const float*)d_in[7];
  const float* b3 = (const float*)d_in[8];

  const int E = in_sizes[2];
  const int Nn = in_sizes[0] / IN_C;
  float* out = (float*)d_out;

  // workspace layout (floats): dinv[N] | wnorm[E] | bufA[N*64] | bufB[N*64]
  float* dinv = (float*)d_ws;
  float* wnorm = dinv + Nn;
  float* bufA = wnorm + E;
  float* bufB = bufA + (size_t)Nn * H1C;

  // --- gcn_norm ---
  k_init_deg<<<blocks_for(Nn, 256), 256, 0, stream>>>(dinv, Nn);
  k_deg_accum<<<blocks_for(E, 256), 256, 0, stream>>>(ei, ew, dinv, E);
  k_dinv<<<blocks_for(Nn, 256), 256, 0, stream>>>(dinv, Nn);
  k_wnorm<<<blocks_for(E, 256), 256, 0, stream>>>(ei, ew, dinv, wnorm, E);

  // --- layer 1: h1 = x @ W1 (WMMA), agg = b1 + D^-1/2 (A+I) D^-1/2 h1 ---
  k_gemm_wmma_f32<IN_C, H1C / 16, false>
      <<<Nn / 16, (H1C / 16) * 32, 0, stream>>>(x, W1, bufA);
  k_init_agg<H1C><<<blocks_for((long long)Nn * H1C, 256), 256, 0, stream>>>(
      bufB, bufA, dinv, b1, Nn);
  k_edge_agg<H1C / 4><<<blocks_for((long long)E * (H1C / 4), 256), 256, 0, stream>>>(
      ei, wnorm, bufA, bufB, E);

  // --- layer 2: h2 = relu(agg1) @ W2 (WMMA) ---
  k_gemm_wmma_f32<H1C, H2C / 16, true>
      <<<Nn / 16, (H2C / 16) * 32, 0, stream>>>(bufB, W2, bufA);
  k_init_agg<H2C><<<blocks_for((long long)Nn * H2C, 256), 256, 0, stream>>>(
      bufB, bufA, dinv, b2, Nn);
  k_edge_agg<H2C / 4><<<blocks_for((long long)E * (H2C / 4), 256), 256, 0, stream>>>(
      ei, wnorm, bufA, bufB, E);

  // --- layer 3: g = relu(agg2) @ W3, out = b3 + D^-1/2 (A+I) D^-1/2 g ---
  k_gemm3_relu<<<blocks_for(Nn, 256), 256, 0, stream>>>(bufB, W3, bufA, Nn);
  k_init_out<<<blocks_for(Nn, 256), 256, 0, stream>>>(out, bufA, dinv, b3, Nn);
  k_edge_agg1<<<blocks_for(E, 256), 256, 0, stream>>>(ei, wnorm, bufA, out, E);
}